// ChamferDistanceLoss_28372553957891
// MI455X (gfx1250) — compile-verified
//
#include <hip/hip_runtime.h>

// ---------------------------------------------------------------------------
// Bidirectional 1-D Chamfer distance (scale-invariant) for MI455X / gfx1250.
//
// Key idea: (b - t)^2 = b^2*1 + (-2b)*t + 1*t^2, so a 16x16 tile of exact
// pairwise squared distances is ONE V_WMMA_F32_16X16X4_F32:
//   A (16x4, rows = bins):    row m = [ b_m^2, -2*b_m, 1, 0 ]
//   B (4x16, cols = targets): col n = [ 1, t_n, t_n^2, 0 ]^T
// Min-reductions are done on the C tile in registers + cross-lane shuffles.
//
// VGPR layout assumptions (CDNA5 ISA 7.12.2, wave32):
//   A 16x4 (f32, 2 VGPRs): M = lane%16, K = (lane/16)*2 + vgpr
//   B 4x16 (f32, 2 VGPRs): N = lane%16, K = (lane/16)*2 + vgpr  (symmetric)
//   C/D 16x16 (8 VGPRs):   N = lane%16, M = vgpr + (lane/16)*8
// ---------------------------------------------------------------------------

typedef __attribute__((ext_vector_type(2))) float v2f;
typedef __attribute__((ext_vector_type(8))) float v8f;

#define NBINS      256
#define NBIN_TILES 16
#define HW         (480 * 640)
#define NGROUPS    (HW / 16)      // 19200 groups of 16 targets (exact)

// Workspace layout (uint32 words in d_ws):
//   [0]       target max  (uint-ordered positive float, atomicMax)
//   [1]       bin max     (uint-ordered positive float, atomicMax)
//   [2]       target-side sum (float, atomicAdd)
//   [3]       pad
//   [4..259]  per-bin min (uint-ordered float, init +inf, atomicMin)

__global__ void k_init(unsigned* __restrict__ ws) {
    int i = threadIdx.x;
    if (i == 0) {
        ws[0] = 0u;
        ws[1] = 0u;
        ((float*)ws)[2] = 0.0f;
    }
    if (i < NBINS) ws[4 + i] = 0x7F800000u;   // +inf
}

__global__ void k_max(const float* __restrict__ tgt,
                      const float* __restrict__ bins,
                      unsigned* __restrict__ ws) {
    int gid    = blockIdx.x * blockDim.x + threadIdx.x;
    int stride = gridDim.x * blockDim.x;
    float m = 0.0f;                            // inputs are in [0.1, 10]
    for (int i = gid; i < HW; i += stride) m = fmaxf(m, tgt[i]);
    #pragma unroll
    for (int s = 16; s >= 1; s >>= 1) m = fmaxf(m, __shfl_xor(m, s, 32));
    if ((threadIdx.x & 31) == 0) atomicMax(&ws[0], __float_as_uint(m));
    if (gid < NBINS) atomicMax(&ws[1], __float_as_uint(bins[gid]));
}

// For each target: min over all 256 bins of (b-t)^2, summed over all targets.
__global__ void k_target_side(const float* __restrict__ tgt,
                              const float* __restrict__ bins,
                              unsigned* __restrict__ ws) {
    const int  lane   = threadIdx.x & 31;
    const int  nWaves = (gridDim.x * blockDim.x) >> 5;
    const int  waveId = (blockIdx.x * blockDim.x + threadIdx.x) >> 5;
    const int  col    = lane & 15;
    const bool lo     = lane < 16;

    const float inv_t = 1.0f / __uint_as_float(ws[0]);
    const float inv_b = 1.0f / __uint_as_float(ws[1]);

    // Prebuild all 16 bin A-tiles (kept in registers: 32 VGPRs).
    v2f A[NBIN_TILES];
    #pragma unroll
    for (int t = 0; t < NBIN_TILES; ++t) {
        float bv = bins[t * 16 + col] * inv_b;
        A[t].x = lo ? bv * bv     : 1.0f;   // K=0 : b^2   | K=2 : 1
        A[t].y = lo ? -2.0f * bv  : 0.0f;   // K=1 : -2b   | K=3 : 0
    }

    float localSum = 0.0f;
    // Wave-uniform grid-stride loop: EXEC stays all-ones around WMMA.
    for (int g = waveId; g < NGROUPS; g += nWaves) {
        float tv = tgt[g * 16 + col] * inv_t;
        v2f B;
        B.x = lo ? 1.0f : tv * tv;          // K=0 : 1     | K=2 : t^2
        B.y = lo ? tv   : 0.0f;             // K=1 : t     | K=3 : 0

        float tmin = 3.0e38f;
        #pragma unroll
        for (int t = 0; t < NBIN_TILES; ++t) {
            v8f c = {};
            c = __builtin_amdgcn_wmma_f32_16x16x4_f32(
                    false, A[t], false, B, (short)0, c, false, false);
            float m = fminf(fminf(fminf(c[0], c[1]), fminf(c[2], c[3])),
                            fminf(fminf(c[4], c[5]), fminf(c[6], c[7])));
            tmin = fminf(tmin, m);
        }
        // Combine row halves: lanes L and L+16 cover rows 0-7 / 8-15.
        tmin = fminf(tmin, __shfl_xor(tmin, 16, 32));
        // Sum the 16 per-target minima of this group across the wave.
        float v = lo ? tmin : 0.0f;
        #pragma unroll
        for (int s = 1; s <= 16; s <<= 1) v += __shfl_xor(v, s, 32);
        localSum += v;                      // fully-reduced, same in all lanes
    }
    if (lane == 0) atomicAdd((float*)&ws[2], localSum);
}

// For each bin: min over all targets of (b-t)^2 (partial mins via atomicMin).
__global__ void k_bin_side(const float* __restrict__ tgt,
                           const float* __restrict__ bins,
                           unsigned* __restrict__ ws) {
    const int  lane    = threadIdx.x & 31;
    const int  nWaves  = (gridDim.x * blockDim.x) >> 5;
    const int  waveId  = (blockIdx.x * blockDim.x + threadIdx.x) >> 5;
    const int  tile    = waveId & (NBIN_TILES - 1);
    const int  chunk   = waveId >> 4;
    const int  nChunks = nWaves >> 4;       // launch keeps nWaves % 16 == 0
    const int  col     = lane & 15;
    const bool lo      = lane < 16;

    const float inv_t = 1.0f / __uint_as_float(ws[0]);
    const float inv_b = 1.0f / __uint_as_float(ws[1]);

    float bv = bins[tile * 16 + col] * inv_b;
    v2f A;
    A.x = lo ? bv * bv    : 1.0f;
    A.y = lo ? -2.0f * bv : 0.0f;

    v8f accum;
    #pragma unroll
    for (int v = 0; v < 8; ++v) accum[v] = 3.0e38f;

    for (int g = chunk; g < NGROUPS; g += nChunks) {
        float tv = tgt[g * 16 + col] * inv_t;
        v2f B;
        B.x = lo ? 1.0f : tv * tv;
        B.y = lo ? tv   : 0.0f;
        v8f c = {};
        c = __builtin_amdgcn_wmma_f32_16x16x4_f32(
                false, A, false, B, (short)0, c, false, false);
        #pragma unroll
        for (int v = 0; v < 8; ++v) accum[v] = fminf(accum[v], c[v]);
    }
    // Min over the 16 columns; xor masks < 16 stay within each lane half.
    #pragma unroll
    for (int s = 1; s <= 8; s <<= 1) {
        #pragma unroll
        for (int v = 0; v < 8; ++v)
            accum[v] = fminf(accum[v], __shfl_xor(accum[v], s, 32));
    }
    if (col == 0) {                         // lanes 0 (rows 0-7) and 16 (8-15)
        int rowBase = tile * 16 + (lo ? 0 : 8);
        #pragma unroll
        for (int v = 0; v < 8; ++v)
            atomicMin(&ws[4 + rowBase + v], __float_as_uint(accum[v]));
    }
}

__global__ void k_final(const unsigned* __restrict__ ws,
                        float* __restrict__ out) {
    __shared__ float red[256];
    int i = threadIdx.x;
    red[i] = __uint_as_float(ws[4 + i]);
    __syncthreads();
    #pragma unroll
    for (int s = 128; s > 0; s >>= 1) {
        if (i < s) red[i] += red[i + s];
        __syncthreads();
    }
    if (i == 0) out[0] = red[0] + ((const float*)ws)[2];
}

extern "C" void kernel_launch(void* const* d_in, const int* in_sizes, int n_in,
                              void* d_out, int out_size, void* d_ws, size_t ws_size,
                              hipStream_t stream) {
    const float* tgt  = (const float*)d_in[0];   // [1,1,480,640] f32
    const float* bins = (const float*)d_in[1];   // [1,256] f32
    float*       out  = (float*)d_out;           // scalar f32
    unsigned*    ws   = (unsigned*)d_ws;         // needs >= 260 words

    k_init       <<<1,   256, 0, stream>>>(ws);
    k_max        <<<256, 256, 0, stream>>>(tgt, bins, ws);
    k_target_side<<<128, 256, 0, stream>>>(tgt, bins, ws);   // 1024 waves
    k_bin_side   <<<64,  256, 0, stream>>>(tgt, bins, ws);   // 512 waves = 16 tiles x 32 chunks
    k_final      <<<1,   256, 0, stream>>>(ws, out);
}